// EdgeDecoder_26474178412607
// MI455X (gfx1250) — compile-verified
//
#include <hip/hip_runtime.h>

// ---------------------------------------------------------------------------
// EdgeDecoder for MI455X (gfx1250, wave32, WMMA).
//   out[e] = b2 + sum_n relu( sum_k zcat[e][k] * w1[k][n] + b1[n] ) * w2[n]
// zcat[e] = concat(z_user[edge_row[e]], z_movie[edge_col[e]])  (1024 f32)
//
// Memory-bound on the 2.05 GB edge gather; fp32 WMMA (16x16x4) keeps full
// precision while the gather dominates (~88us at 23.3 TB/s).
// ---------------------------------------------------------------------------

typedef __attribute__((ext_vector_type(2))) float v2f;
typedef __attribute__((ext_vector_type(8))) float v8f;

namespace {
constexpr int kH      = 512;    // hidden dim
constexpr int kTwoH   = 1024;   // concat dim (K of the GEMM)
constexpr int kMTile  = 32;     // edges per workgroup
constexpr int kRowStr = 1028;   // padded LDS row stride in floats (1028%64==4 -> conflict-free A reads)
constexpr int kKSteps = kTwoH / 4;  // 256 WMMA K-steps
}

// Pack w1 [1024,512] row-major into B-fragment order for V_WMMA_F32_16X16X4_F32:
// for k-step g, half kh, column n: w1p[(g*2+kh)*512+n] = (w1[4g+2kh][n], w1[4g+2kh+1][n])
// Lane (kh, laneM) of a wave then loads its (VGPR0, VGPR1) B pair as one b64.
__global__ void repack_w1_kernel(const float* __restrict__ w1,
                                 float2* __restrict__ w1p) {
    int t = blockIdx.x * blockDim.x + threadIdx.x;
    if (t >= kKSteps * 2 * kH) return;
    int n    = t & (kH - 1);
    int gk   = t >> 9;          // g*2 + kh
    int kh   = gk & 1;
    int g    = gk >> 1;
    int krow = 4 * g + 2 * kh;
    float2 v;
    v.x = w1[krow * kH + n];
    v.y = w1[(krow + 1) * kH + n];
    w1p[t] = v;
}

__global__ __launch_bounds__(256) void edge_decoder_kernel(
    const float* __restrict__ z_user,  const float* __restrict__ z_movie,
    const int*   __restrict__ edge_row, const int* __restrict__ edge_col,
    const float2* __restrict__ w1p,
    const float* __restrict__ bias1, const float* __restrict__ w2v,
    const float* __restrict__ bias2,
    float* __restrict__ out, int E) {
    extern __shared__ float smem[];                 // [32][1028] A-stage + [32] acc
    float* acc = smem + kMTile * kRowStr;

    const int tid   = threadIdx.x;                  // 0..255 (8 waves of 32)
    const int wave  = tid >> 5;
    const int lane  = tid & 31;
    const int laneM = lane & 15;
    const int kh    = lane >> 4;                    // K-half within fragment
    const int edge0 = blockIdx.x * kMTile;

    // ---- stage 32 gathered concat rows into LDS (8 threads per row) ----
    {
        int r  = tid >> 3;                          // row 0..31
        int c  = tid & 7;
        int e  = edge0 + r;
        int ec = e < E ? e : (E - 1);
        const float4* zu = (const float4*)(z_user  + (size_t)edge_row[ec] * kH);
        const float4* zm = (const float4*)(z_movie + (size_t)edge_col[ec] * kH);
        float4* du = (float4*)(smem + r * kRowStr);         // first 512 floats
        float4* dm = (float4*)(smem + r * kRowStr + kH);    // next 512 floats
#pragma unroll
        for (int i = 0; i < 16; ++i) du[c + 8 * i] = zu[c + 8 * i];
#pragma unroll
        for (int i = 0; i < 16; ++i) dm[c + 8 * i] = zm[c + 8 * i];
    }
    if (tid < kMTile) acc[tid] = 0.0f;
    __syncthreads();

    // ---- GEMM: C[32x64 per wave] = A[32x1024] x w1[1024 x (wave's 64 cols)] ----
    // A frag (16x4 f32): lane(laneM,kh) holds A[laneM][k+2kh], A[laneM][k+2kh+1]
    const float*  aBase0 = smem + laneM * kRowStr + 2 * kh;        // M-tile 0
    const float*  aBase1 = aBase0 + 16 * kRowStr;                  // M-tile 1
    const float2* bBase  = w1p + kh * kH + wave * 64 + laneM;      // packed B

    v8f cacc[2][4];
#pragma unroll
    for (int mt = 0; mt < 2; ++mt)
#pragma unroll
        for (int nt = 0; nt < 4; ++nt)
#pragma unroll
            for (int j = 0; j < 8; ++j) cacc[mt][nt][j] = 0.0f;

#pragma unroll 4
    for (int g = 0; g < kKSteps; ++g) {
        v2f a0 = *(const v2f*)(aBase0 + 4 * g);     // ds_load_b64, bank-conflict-free
        v2f a1 = *(const v2f*)(aBase1 + 4 * g);
        const float2* bp = bBase + g * (2 * kH);    // +1024 float2 per k-step
#pragma unroll
        for (int nt = 0; nt < 4; ++nt) {
            v2f b = *(const v2f*)(bp + nt * 16);
            cacc[0][nt] = __builtin_amdgcn_wmma_f32_16x16x4_f32(
                false, a0, false, b, (short)0, cacc[0][nt], false, false);
            cacc[1][nt] = __builtin_amdgcn_wmma_f32_16x16x4_f32(
                false, a1, false, b, (short)0, cacc[1][nt], false, false);
        }
    }

    // ---- epilogue: relu(+bias) dot w2, butterfly-reduce over N, LDS accumulate ----
    const int nb = wave * 64 + laneM;               // lane's N coordinate base
    float bia[4], wv[4];
#pragma unroll
    for (int nt = 0; nt < 4; ++nt) {
        bia[nt] = bias1[nb + nt * 16];
        wv[nt]  = w2v[nb + nt * 16];
    }

#pragma unroll
    for (int mt = 0; mt < 2; ++mt) {
#pragma unroll
        for (int i = 0; i < 8; ++i) {
            // C layout: VGPR i -> lanes 0-15: (M=i, N=laneM); lanes 16-31: (M=i+8, N=laneM)
            float s = 0.0f;
#pragma unroll
            for (int nt = 0; nt < 4; ++nt) {
                float v = cacc[mt][nt][i] + bia[nt];
                v = v > 0.0f ? v : 0.0f;
                s += v * wv[nt];
            }
            s += __shfl_xor(s, 1);
            s += __shfl_xor(s, 2);
            s += __shfl_xor(s, 4);
            s += __shfl_xor(s, 8);                  // 16-lane group sum
            if (laneM == 0) {
                int m = mt * 16 + kh * 8 + i;
                atomicAdd(&acc[m], s);              // ds_add_f32
            }
        }
    }
    __syncthreads();

    if (tid < kMTile) {
        int e = edge0 + tid;
        if (e < E) out[e] = acc[tid] + bias2[0];
    }
}

extern "C" void kernel_launch(void* const* d_in, const int* in_sizes, int n_in,
                              void* d_out, int out_size, void* d_ws, size_t ws_size,
                              hipStream_t stream) {
    const float* z_user   = (const float*)d_in[0];
    const float* z_movie  = (const float*)d_in[1];
    const int*   edge_row = (const int*)d_in[2];
    const int*   edge_col = (const int*)d_in[3];
    const float* w1       = (const float*)d_in[4];
    const float* bias1    = (const float*)d_in[5];
    const float* w2v      = (const float*)d_in[6];
    const float* bias2    = (const float*)d_in[7];
    float* out = (float*)d_out;
    const int E = in_sizes[2];

    // Scratch: packed w1 fragments (2 MB), rebuilt every launch (deterministic).
    float2* w1p = (float2*)d_ws;
    const int total = kKSteps * 2 * kH;             // 262144 float2
    repack_w1_kernel<<<(total + 255) / 256, 256, 0, stream>>>(w1, w1p);

    const int nblk = (E + kMTile - 1) / kMTile;     // 15625 for E=500000
    const size_t smem_bytes = (size_t)(kMTile * kRowStr + kMTile) * sizeof(float);
    edge_decoder_kernel<<<nblk, 256, smem_bytes, stream>>>(
        z_user, z_movie, edge_row, edge_col, w1p, bias1, w2v, bias2, out, E);
}